// SAGE_Net_81432579932422
// MI455X (gfx1250) — compile-verified
//
#include <hip/hip_runtime.h>
#include <math.h>

typedef float v2f __attribute__((ext_vector_type(2)));
typedef float v8f __attribute__((ext_vector_type(8)));

#define D_FEAT 256

// ---------------------------------------------------------------- fill
__global__ void fill_f32(float* __restrict__ p, long n, float v) {
    long i = (long)blockIdx.x * blockDim.x + threadIdx.x;
    if (i < n) p[i] = v;
}

// ---------------------------------------------------------------- degree
__global__ void deg_count(const int* __restrict__ edges, float* __restrict__ deg,
                          int E) {
    int e = blockIdx.x * blockDim.x + threadIdx.x;
    if (e < E) atomicAdd(&deg[edges[E + e]], 1.0f);   // dst row
}

__global__ void deg_invert(float* __restrict__ deg, int N) {
    int i = blockIdx.x * blockDim.x + threadIdx.x;
    if (i < N) deg[i] = 1.0f / fmaxf(deg[i], 1.0f);
}

// ---------------------------------------------------------------- scatter-add aggregation
// 64 threads per edge, 4 consecutive floats per thread (coalesced 1KB row segments;
// x/h live in the 192MB L2, atomics resolve at L2).
__global__ void aggregate(const float* __restrict__ feat,
                          const int* __restrict__ edges,
                          float* __restrict__ agg, int E) {
    long t  = (long)blockIdx.x * blockDim.x + threadIdx.x;
    int  e  = (int)(t >> 6);
    int  c  = (int)(t & 63) << 2;
    if (e >= E) return;
    int src = edges[e];
    int dst = edges[E + e];
    const float4 v = *(const float4*)(feat + (size_t)src * D_FEAT + c);
    float* o = agg + (size_t)dst * D_FEAT + c;
    atomicAdd(o + 0, v.x);
    atomicAdd(o + 1, v.y);
    atomicAdd(o + 2, v.z);
    atomicAdd(o + 3, v.w);
}

// ---------------------------------------------------------------- fused SAGE linear
// Out[m,n] = act( (deg_inv[m]*Agg[m,:]) @ Wl[:,n] + X[m,:] @ Wr[:,n] + bias[n] )
//
// One wave computes a 16x64 output slab: 4 n-tiles, 4 independent f32
// accumulators, fp32 WMMA 16x16x4, K fully unrolled at compile time so every
// global load uses a fixed per-lane base pointer + constant immediate offset
// (no address arithmetic in the loop body).
//
// No EXEC-divergent guards in the hot loop: weight-column indices are clamped
// to [0, C-1] (a junk B column only pollutes an output column we never store);
// the C-bound is enforced only at the epilogue store.
template <int K, int C, int RELU>
__global__ void sage_gemm_wmma(const float* __restrict__ Agg,
                               const float* __restrict__ X,
                               const float* __restrict__ Wl,
                               const float* __restrict__ Wr,
                               const float* __restrict__ bias,
                               const float* __restrict__ dinv,
                               float* __restrict__ Out) {
    const int lane = threadIdx.x;        // 0..31 (wave32)
    const int half = lane >> 4;          // 0: K pair {k,k+1}; 1: {k+2,k+3}
    const int lm   = lane & 15;
    const int m0   = blockIdx.x * 16;
    const int nb   = threadIdx.y * 64;   // this wave's 64-col stripe

    const int rowA    = m0 + lm;         // this lane's A/X row (M dim)
    const float scale = dinv[rowA];

    // fixed per-lane base pointers (computed once; loop offsets are immediates)
    const float* ap = Agg + (size_t)rowA * K + 2 * half;
    const float* xp = X   + (size_t)rowA * K + 2 * half;

    const float* pwl[4];
    const float* pwr[4];
#pragma unroll
    for (int t = 0; t < 4; ++t) {
        int nc  = nb + 16 * t + lm;
        int ncc = (nc < C) ? nc : (C - 1);          // clamped, always legal
        pwl[t] = Wl + (size_t)(2 * half) * C + ncc;
        pwr[t] = Wr + (size_t)(2 * half) * C + ncc;
    }

    v8f acc[4] = {};
#pragma unroll
    for (int k = 0; k < K; k += 4) {
        // A fragments (16x4 f32): vgpr0 = K=k+2*half, vgpr1 = K=k+1+2*half
        v2f a, xa;
        a.x  = ap[k]     * scale;
        a.y  = ap[k + 1] * scale;
        xa.x = xp[k];
        xa.y = xp[k + 1];

#pragma unroll
        for (int t = 0; t < 4; ++t) {
            v2f bl, br;
            bl.x = pwl[t][k * C];          // row k+2*half   (immediate offset)
            bl.y = pwl[t][k * C + C];      // row k+2*half+1
            br.x = pwr[t][k * C];
            br.y = pwr[t][k * C + C];
            acc[t] = __builtin_amdgcn_wmma_f32_16x16x4_f32(
                false, a, false, bl, (short)0, acc[t], false, false);
            acc[t] = __builtin_amdgcn_wmma_f32_16x16x4_f32(
                false, xa, false, br, (short)0, acc[t], false, false);
        }
    }

    // epilogue: bias (+ReLU) and bounds-guarded store
#pragma unroll
    for (int t = 0; t < 4; ++t) {
        const int nc = nb + 16 * t + lm;
        if (nc < C) {
            const float bv = bias[nc];
#pragma unroll
            for (int r = 0; r < 8; ++r) {
                const int row = m0 + r + 8 * half;   // C/D layout: vgpr r -> M=r(+8)
                float v = acc[t][r] + bv;
                if (RELU) v = fmaxf(v, 0.0f);
                Out[(size_t)row * C + nc] = v;
            }
        }
    }
}

// ---------------------------------------------------------------- log_softmax over 40 cols
// One wave per row; lane covers cols {lane, lane+32}; shfl_xor tree reductions.
__global__ void log_softmax40(float* __restrict__ out, int N) {
    const int row  = blockIdx.x * (blockDim.x >> 5) + (threadIdx.x >> 5);
    const int lane = threadIdx.x & 31;
    if (row >= N) return;
    float* p = out + (size_t)row * 40;

    const float v0 = p[lane];                                   // lane < 40 always
    const bool  h1 = (lane + 32) < 40;
    const float v1 = h1 ? p[lane + 32] : -INFINITY;

    float m = fmaxf(v0, v1);
#pragma unroll
    for (int off = 16; off; off >>= 1) m = fmaxf(m, __shfl_xor(m, off, 32));

    float s = expf(v0 - m) + (h1 ? expf(v1 - m) : 0.0f);
#pragma unroll
    for (int off = 16; off; off >>= 1) s += __shfl_xor(s, off, 32);

    const float lse = m + logf(s);
    p[lane] = v0 - lse;
    if (h1) p[lane + 32] = v1 - lse;
}

// ---------------------------------------------------------------- launcher
extern "C" void kernel_launch(void* const* d_in, const int* in_sizes, int n_in,
                              void* d_out, int out_size, void* d_ws, size_t ws_size,
                              hipStream_t stream) {
    const float* x   = (const float*)d_in[0];
    const int*   ei  = (const int*)  d_in[1];   // [2, E] (jax default int32)
    const float* W1l = (const float*)d_in[2];
    const float* W1r = (const float*)d_in[3];
    const float* b1  = (const float*)d_in[4];
    const float* W2l = (const float*)d_in[5];
    const float* W2r = (const float*)d_in[6];
    const float* b2  = (const float*)d_in[7];
    float*       out = (float*)d_out;

    const int N = in_sizes[0] / D_FEAT;    // 100000
    const int E = in_sizes[1] / 2;         // 3200000

    // workspace: deg_inv [N] | agg [N*256] | h [N*256]  (~205 MB)
    float* deg = (float*)d_ws;
    float* agg = deg + (((size_t)N + 255) & ~(size_t)255);
    float* h   = agg + (size_t)N * D_FEAT;

    const long nAgg = (long)N * D_FEAT;

    // degree + inverse
    fill_f32<<<(N + 255) / 256, 256, 0, stream>>>(deg, N, 0.0f);
    deg_count<<<(E + 255) / 256, 256, 0, stream>>>(ei, deg, E);
    deg_invert<<<(N + 255) / 256, 256, 0, stream>>>(deg, N);

    // layer 1: aggregate x, then fused WMMA linear + ReLU -> h
    fill_f32<<<(int)((nAgg + 255) / 256), 256, 0, stream>>>(agg, nAgg, 0.0f);
    aggregate<<<(int)(((long)E * 64 + 255) / 256), 256, 0, stream>>>(x, ei, agg, E);
    {
        // 4 waves/block, each wave owns a 64-col stripe -> block = 16x256 slab
        dim3 grid(N / 16, 1), block(32, 4);
        sage_gemm_wmma<256, 256, 1>
            <<<grid, block, 0, stream>>>(agg, x, W1l, W1r, b1, deg, h);
    }

    // layer 2: aggregate h, fused WMMA linear -> out (C = 40 fits one 64-col stripe)
    fill_f32<<<(int)((nAgg + 255) / 256), 256, 0, stream>>>(agg, nAgg, 0.0f);
    aggregate<<<(int)(((long)E * 64 + 255) / 256), 256, 0, stream>>>(h, ei, agg, E);
    {
        dim3 grid(N / 16, 1), block(32, 1);
        sage_gemm_wmma<256, 40, 0>
            <<<grid, block, 0, stream>>>(agg, h, W2l, W2r, b2, deg, out);
    }

    // row-wise log_softmax in place
    log_softmax40<<<(N + 7) / 8, 256, 0, stream>>>(out, N);
}